// CoordinateSparseAttention_20323785244775
// MI455X (gfx1250) — compile-verified
//
#include <hip/hip_runtime.h>

typedef __attribute__((ext_vector_type(2))) float v2f;
typedef __attribute__((ext_vector_type(8))) float v8f;

#define C_CH   64
#define H_DIM  96
#define W_DIM  96
#define N_PIX  (H_DIM * W_DIM)   // 9216
#define B_SZ   4
#define O_CH   192               // 3*C
#define KNN    8

// ---------------------------------------------------------------------------
// Kernel 1: exact top-8 nearest-neighbor indices on the uniform grid.
// d2 is proportional to di*di+dj*dj (integers), so we select with the exact
// integer key (d2<<14 | index), matching jax.lax.top_k's value-then-lowest-
// index ordering. The 8 nearest always lie within a 5x5 window (worst corner
// case: max selected d2 = 5 < 9 = closest candidate outside the window).
// ---------------------------------------------------------------------------
__global__ __launch_bounds__(256) void knn_topk_kernel(int* __restrict__ topk) {
    int n = blockIdx.x * blockDim.x + threadIdx.x;
    if (n >= N_PIX) return;
    int qi = n / W_DIM, qj = n % W_DIM;
    unsigned best[KNN];
#pragma unroll
    for (int j = 0; j < KNN; ++j) best[j] = 0xFFFFFFFFu;
    for (int di = -2; di <= 2; ++di) {
        int mi = qi + di;
        if (mi < 0 || mi >= H_DIM) continue;
        for (int dj = -2; dj <= 2; ++dj) {
            int mj = qj + dj;
            if (mj < 0 || mj >= W_DIM) continue;
            int m = mi * W_DIM + mj;
            unsigned key = ((unsigned)(di * di + dj * dj) << 14) | (unsigned)m;
            if (key < best[KNN - 1]) {
                int p = KNN - 1;
                while (p > 0 && best[p - 1] > key) { best[p] = best[p - 1]; --p; }
                best[p] = key;
            }
        }
    }
#pragma unroll
    for (int j = 0; j < KNN; ++j) topk[n * KNN + j] = (int)(best[j] & 0x3FFFu);
}

// ---------------------------------------------------------------------------
// Kernel 2: QKV projection GEMM via V_WMMA_F32_16X16X4_F32 (f32-native WMMA).
//   qkv[b,o,n] = sum_c w[o,c] * x[b,c,n] + bias[o]
// One wave computes a 16(o) x 64(n) strip: A (weights) preloaded into 16 v2f
// registers and reused across 4 column tiles; 16 chained WMMAs per tile (K=64).
// VGPR layouts per CDNA5 ISA 7.12.2:
//   A  lane l: row (l&15), k = 2*(l>>4) + {0,1} within each K=4 step
//   B  lane l: col (l&15), same k pair
//   D  vgpr r, lane l: row r + 8*(l>>4), col (l&15)
// ---------------------------------------------------------------------------
__global__ __launch_bounds__(128) void qkv_gemm_wmma_kernel(
    const float* __restrict__ x,      // [B, C, N]
    const float* __restrict__ w,      // [3C, C] row-major
    const float* __restrict__ bias,   // [3C]
    float* __restrict__ qkv)          // [B, 3C, N]
{
    const int lane = threadIdx.x & 31;
    const int wv   = threadIdx.x >> 5;
    const int m16  = lane & 15;
    const int half = lane >> 4;        // 0 or 1
    const int kb   = half * 2;         // k-pair base within each K=4 step
    const int b    = blockIdx.z;
    const int o0   = blockIdx.y * 16;
    const int n0   = blockIdx.x * 256 + wv * 64;

    // Preload A: 16 o-rows x 64 K values -> 16 v2f per lane, reused 4x.
    const float* wrow = w + (o0 + m16) * C_CH;
    v2f a[16];
#pragma unroll
    for (int kk = 0; kk < 16; ++kk) {
        a[kk].x = wrow[4 * kk + kb];
        a[kk].y = wrow[4 * kk + kb + 1];
    }

    const float* xb = x + (size_t)b * C_CH * N_PIX;
    float* qb       = qkv + (size_t)b * O_CH * N_PIX;

#pragma unroll
    for (int t = 0; t < 4; ++t) {
        const int nn = n0 + t * 16 + m16;
        v8f acc = {0.f, 0.f, 0.f, 0.f, 0.f, 0.f, 0.f, 0.f};
#pragma unroll
        for (int kk = 0; kk < 16; ++kk) {
            const int c0 = 4 * kk + kb;
            v2f bb;
            bb.x = xb[(size_t)c0 * N_PIX + nn];
            bb.y = xb[(size_t)(c0 + 1) * N_PIX + nn];
            // 8 args: (neg_a, A, neg_b, B, c_mod, C, reuse_a, reuse_b)
            acc = __builtin_amdgcn_wmma_f32_16x16x4_f32(
                false, a[kk], false, bb, (short)0, acc, false, false);
        }
#pragma unroll
        for (int r = 0; r < 8; ++r) {
            const int o = o0 + r + 8 * half;
            qb[(size_t)o * N_PIX + nn] = acc[r] + bias[o];
        }
    }
}

// ---------------------------------------------------------------------------
// Kernel 3: 8-neighbor sparse attention, one wave32 per (b, n).
// Lane l owns channels {l, l+32}. Dot products reduced with cross-lane
// shuffles (wave32-native). Softmax replicated per lane (all lanes identical).
// ---------------------------------------------------------------------------
__global__ __launch_bounds__(256) void attn_kernel(
    const float* __restrict__ qkv,    // [B, 3C, N]
    const int*   __restrict__ topk,   // [N, 8]
    const float* __restrict__ pos,    // [C]
    float*       __restrict__ out)    // [B, C, N]
{
    const int lane = threadIdx.x & 31;
    const int wv   = threadIdx.x >> 5;
    const int idx  = blockIdx.x * 8 + wv;   // in [0, B*N), exact multiple of 8
    const int b = idx / N_PIX;
    const int n = idx % N_PIX;

    const float* qb = qkv + (size_t)b * O_CH * N_PIX;
    const float q0 = qb[(size_t)lane * N_PIX + n];
    const float q1 = qb[(size_t)(lane + 32) * N_PIX + n];

    float logit[KNN], v0[KNN], v1[KNN];
#pragma unroll
    for (int j = 0; j < KNN; ++j) {
        const int m = topk[n * KNN + j];
        const float k0 = qb[(size_t)(64 + lane) * N_PIX + m];
        const float k1 = qb[(size_t)(96 + lane) * N_PIX + m];
        v0[j] = qb[(size_t)(128 + lane) * N_PIX + m];
        v1[j] = qb[(size_t)(160 + lane) * N_PIX + m];
        float p = q0 * k0 + q1 * k1;
#pragma unroll
        for (int off = 16; off; off >>= 1) p += __shfl_xor(p, off, 32);
        logit[j] = p * 0.125f;   // 1/sqrt(64)
    }

    float mx = logit[0];
#pragma unroll
    for (int j = 1; j < KNN; ++j) mx = fmaxf(mx, logit[j]);
    float e[KNN], s = 0.f;
#pragma unroll
    for (int j = 0; j < KNN; ++j) { e[j] = __expf(logit[j] - mx); s += e[j]; }
    const float inv = 1.f / s;

    float o0 = 0.f, o1 = 0.f;
#pragma unroll
    for (int j = 0; j < KNN; ++j) {
        const float p = e[j] * inv;
        o0 += p * v0[j];
        o1 += p * v1[j];
    }
    out[((size_t)b * C_CH + lane) * N_PIX + n]        = o0 + pos[lane];
    out[((size_t)b * C_CH + lane + 32) * N_PIX + n]   = o1 + pos[lane + 32];
}

// ---------------------------------------------------------------------------
extern "C" void kernel_launch(void* const* d_in, const int* in_sizes, int n_in,
                              void* d_out, int out_size, void* d_ws, size_t ws_size,
                              hipStream_t stream) {
    (void)in_sizes; (void)n_in; (void)out_size; (void)ws_size;
    const float* x    = (const float*)d_in[0];   // [4,64,96,96]
    const float* wq   = (const float*)d_in[1];   // [192,64]
    const float* bq   = (const float*)d_in[2];   // [192]
    const float* pos  = (const float*)d_in[3];   // [64]
    float* out = (float*)d_out;                  // [4,64,96,96]

    // Workspace layout: topk (288 KB, 256B-aligned) then qkv scratch (28.3 MB).
    int*   topk = (int*)d_ws;
    size_t qkv_off = ((size_t)N_PIX * KNN * sizeof(int) + 255) & ~(size_t)255;
    float* qkv = (float*)((char*)d_ws + qkv_off);

    knn_topk_kernel<<<(N_PIX + 255) / 256, 256, 0, stream>>>(topk);

    dim3 g(N_PIX / 256, O_CH / 16, B_SZ);        // 36 x 12 x 4
    qkv_gemm_wmma_kernel<<<g, 128, 0, stream>>>(x, wq, bq, qkv);

    attn_kernel<<<(B_SZ * N_PIX) / 8, 256, 0, stream>>>(qkv, topk, pos, out);
}